// DIoULoss_5360119186047
// MI455X (gfx1250) — compile-verified
//
#include <hip/hip_runtime.h>
#include <hip/hip_bf16.h>

#define EPS 1e-7f
#define MAX_TGT 4096
#define BLOCK_THREADS 128   // 4 waves of 32 on CDNA5

typedef unsigned int u32x4 __attribute__((ext_vector_type(4)));
typedef int          i32x8 __attribute__((ext_vector_type(8)));
typedef int          i32x4 __attribute__((ext_vector_type(4)));
typedef float        v2f   __attribute__((ext_vector_type(2)));
typedef float        v8f   __attribute__((ext_vector_type(8)));

// ---------------------------------------------------------------------------
// Stage 1: per-pred argmax over all targets (pairwise IoU, xyxy semantics),
// then DIoU (cxcywh semantics) for the matched pair, wave-reduced partial sums.
// Targets are staged into LDS with the Tensor Data Mover (CDNA5 TDM).
// ---------------------------------------------------------------------------
__global__ __launch_bounds__(BLOCK_THREADS) void diou_stage1(
    const float4* __restrict__ pred,
    const float4* __restrict__ tgt,
    int n_pred, int n_tgt,
    float* __restrict__ ws)
{
    __shared__ float4 s_tgt[MAX_TGT];   // 64 KB of the 320 KB/WGP LDS

    const int tid = threadIdx.x;
    const int nt  = (n_tgt < MAX_TGT) ? n_tgt : MAX_TGT;

#if __has_builtin(__builtin_amdgcn_tensor_load_to_lds)
    // --- TDM: DMA all target boxes (nt*4 f32, contiguous) into LDS ----------
    if (tid < 32) {   // one wave issues the descriptor
        unsigned long long ga = (unsigned long long)(const void*)tgt;
        unsigned lds_off = (unsigned)(unsigned long long)(void*)s_tgt; // low 32b of flat LDS addr = LDS offset
        unsigned elems   = (unsigned)nt * 4u;       // f32 elements (fits 16-bit tile_dim0)

        u32x4 g0;
        g0[0] = 1u;                                  // count=1 (valid), user mode
        g0[1] = lds_off;                             // lds_addr (bytes)
        g0[2] = (unsigned)ga;                        // global_addr[31:0]
        g0[3] = (unsigned)((ga >> 32) & 0x01FFFFFFull) | 0x80000000u; // addr[56:32] | type=2

        i32x8 g1;
        g1[0] = (int)(2u << 16);                     // workgroup_mask=0, data_size=2 (4 bytes)
        g1[1] = (int)((elems & 0xFFFFu) << 16);      // tensor_dim0[15:0]
        g1[2] = (int)(((elems >> 16) & 0xFFFFu) | (1u << 16)); // dim0[31:16] | tensor_dim1=1
        g1[3] = (int)((elems & 0xFFFFu) << 16);      // tensor_dim1 hi=0 | tile_dim0=elems
        g1[4] = 0;                                   // tile_dim1=0 (unused), tile_dim2=0
        g1[5] = (int)elems;                          // tensor_dim0_stride[31:0]
        g1[6] = 0;                                   // stride hi, dim1_stride lo
        g1[7] = 0;

        i32x4 gz = {0, 0, 0, 0};
#if defined(__clang_major__) && (__clang_major__ >= 23)
        i32x8 gz8 = {0, 0, 0, 0, 0, 0, 0, 0};
        __builtin_amdgcn_tensor_load_to_lds(g0, g1, gz, gz, gz8, 0);
#else
        __builtin_amdgcn_tensor_load_to_lds(g0, g1, gz, gz, 0);
#endif
        __builtin_amdgcn_s_wait_tensorcnt(0);        // TENSORcnt == 0 -> tile resident
    }
#else
    for (int j = tid; j < nt; j += BLOCK_THREADS) s_tgt[j] = tgt[j];
#endif
    __syncthreads();

    // --- per-thread pred row -------------------------------------------------
    int i = blockIdx.x * BLOCK_THREADS + tid;
    int ic = (i < n_pred) ? i : (n_pred - 1);
    float4 p = pred[ic];

    // pairwise IoU treats columns as x1,y1,x2,y2
    const float pa = (p.z - p.x) * (p.w - p.y);

    float best  = -__builtin_inff();
    int   bestj = 0;
#pragma unroll 4
    for (int j = 0; j < nt; ++j) {
        float4 t = s_tgt[j];                         // uniform addr -> LDS broadcast
        float ltx = fmaxf(p.x, t.x);
        float lty = fmaxf(p.y, t.y);
        float rbx = fminf(p.z, t.z);
        float rby = fminf(p.w, t.w);
        float w   = fmaxf(rbx - ltx, 0.0f);
        float h   = fmaxf(rby - lty, 0.0f);
        float inter = w * h;
        float ta  = (t.z - t.x) * (t.w - t.y);
        float iou = inter / (pa + ta - inter + EPS);
        if (iou > best) { best = iou; bestj = j; }   // first-max, matches jnp.argmax
    }

    // DIoU treats columns as cx,cy,w,h
    float4 t = s_tgt[bestj];
    float phw = p.z * 0.5f, phh = p.w * 0.5f;
    float thw = t.z * 0.5f, thh = t.w * 0.5f;
    float pltx = p.x - phw, plty = p.y - phh, prbx = p.x + phw, prby = p.y + phh;
    float tltx = t.x - thw, tlty = t.y - thh, trbx = t.x + thw, trby = t.y + thh;

    float iw = fmaxf(fminf(prbx, trbx) - fmaxf(pltx, tltx), 0.0f);
    float ih = fmaxf(fminf(prby, trby) - fmaxf(plty, tlty), 0.0f);
    float inter = iw * ih;
    float iou   = inter / (p.z * p.w + t.z * t.w - inter + EPS);

    float dx = p.x - t.x, dy = p.y - t.y;
    float cdist = dx * dx + dy * dy;

    float ew = fmaxf(prbx, trbx) - fminf(pltx, tltx);
    float eh = fmaxf(prby, trby) - fminf(plty, tlty);
    float diag = ew * ew + eh * eh;

    float diou = iou - cdist / (diag + EPS);
    if (i >= n_pred) diou = 0.0f;

    // wave32 tree reduction; one partial per wave
#pragma unroll
    for (int off = 16; off > 0; off >>= 1)
        diou += __shfl_xor(diou, off, 32);
    if ((tid & 31) == 0)
        ws[blockIdx.x * (BLOCK_THREADS / 32) + (tid >> 5)] = diou;
}

// ---------------------------------------------------------------------------
// Stage 2: single wave. Sum the per-wave partials with chained
// V_WMMA_F32_16X16X4_F32 ops:  D = ones(16x4) x B(4x16) + C.
// Since A is all-ones, sum(all 256 entries of D) == 16 * sum(all B entries),
// independent of the exact lane<->(k,n) mapping. 64 partials per WMMA.
// Tail handling is branchless (clamped index + v_cndmask select) so EXEC
// stays all-1s around every WMMA without save/restore churn.
// ---------------------------------------------------------------------------
__global__ __launch_bounds__(32) void diou_stage2(
    const float* __restrict__ ws, float* __restrict__ out,
    int nparts, float inv_n)
{
    const int lane = threadIdx.x;

    v2f a; a[0] = 1.0f; a[1] = 1.0f;   // ones A-matrix (16x4 f32 across the wave)
    v8f c = {};                        // f32 accumulator (16x16)

    const int ngroups = (nparts + 63) >> 6;
    const int last    = nparts - 1;
    for (int g = 0; g < ngroups; ++g) {
        int i0 = (g << 6) + lane;
        int i1 = i0 + 32;
        int c0 = (i0 <= last) ? i0 : last;           // clamp, load unconditionally
        int c1 = (i1 <= last) ? i1 : last;
        float w0 = ws[c0];
        float w1 = ws[c1];
        v2f b;
        b[0] = (i0 <= last) ? w0 : 0.0f;             // v_cndmask, no EXEC change
        b[1] = (i1 <= last) ? w1 : 0.0f;
        c = __builtin_amdgcn_wmma_f32_16x16x4_f32(
                /*neg_a=*/false, a, /*neg_b=*/false, b,
                /*c_mod=*/(short)0, c, /*reuse_a=*/false, /*reuse_b=*/false);
    }

    float s = c[0] + c[1] + c[2] + c[3] + c[4] + c[5] + c[6] + c[7];
#pragma unroll
    for (int off = 16; off > 0; off >>= 1)
        s += __shfl_xor(s, off, 32);

    if (lane == 0)
        out[0] = 1.0f - (s * (1.0f / 16.0f)) * inv_n;  // 1 - mean(diou)
}

// ---------------------------------------------------------------------------
extern "C" void kernel_launch(void* const* d_in, const int* in_sizes, int n_in,
                              void* d_out, int out_size, void* d_ws, size_t ws_size,
                              hipStream_t stream) {
    const float4* pred = (const float4*)d_in[0];
    const float4* tgt  = (const float4*)d_in[1];
    const int n_pred = in_sizes[0] / 4;   // 8192
    const int n_tgt  = in_sizes[1] / 4;   // 4096
    float* ws  = (float*)d_ws;
    float* out = (float*)d_out;

    const int blocks = (n_pred + BLOCK_THREADS - 1) / BLOCK_THREADS;  // 64
    diou_stage1<<<blocks, BLOCK_THREADS, 0, stream>>>(pred, tgt, n_pred, n_tgt, ws);

    const int nparts = blocks * (BLOCK_THREADS / 32);                 // 256
    diou_stage2<<<1, 32, 0, stream>>>(ws, out, nparts, 1.0f / (float)n_pred);
}